// PScanTriton_68728066670961
// MI455X (gfx1250) — compile-verified
//
#include <hip/hip_runtime.h>

// Problem constants (match reference): B=8, L=1024, H=8, D=32
#define B_ 8
#define L_ 1024
#define H_ 8
#define D_ 32
#define CH 32                      // chunk length (time steps per block)
#define NC (L_ / CH)               // 32 chunks per sequence
#define NSEQ (B_ * H_)             // 64 independent (b,h) chains
#define NCHUNKS (NSEQ * NC)        // 2048 total chunks
#define LDP 33                     // padded LDS row stride (conflict-free columns)
#define ASTRIDE (H_ * D_ * D_)     // element stride between A_t tiles of one (b,h)

typedef float v2f __attribute__((ext_vector_type(2)));
typedef float v8f __attribute__((ext_vector_type(8)));
typedef unsigned int u32x4 __attribute__((ext_vector_type(4)));
typedef int i32x4 __attribute__((ext_vector_type(4)));
typedef int i32x8 __attribute__((ext_vector_type(8)));

#if defined(__gfx1250__) && __has_builtin(__builtin_amdgcn_tensor_load_to_lds)
#define USE_TDM 1
#else
#define USE_TDM 0
#endif

__device__ __forceinline__ v8f wmma_f32_16x16x4(v2f a, v2f b, v8f c) {
  // D = A(16x4) * B(4x16) + C(16x16), fp32 WMMA (CDNA5)
  return __builtin_amdgcn_wmma_f32_16x16x4_f32(
      /*neg_a=*/false, a, /*neg_b=*/false, b,
      /*c_mod=*/(short)0, c, /*reuse_a=*/false, /*reuse_b=*/false);
}

#if USE_TDM
// Issue a TDM 2D tile load: rows x 32 f32 tile, contiguous in global memory
// (row stride 32 floats), into LDS with 1 DWORD of padding inserted every
// 32 DWORDs (pad_interval code 4 = 32 DW, pad_amount code 0 = 1 DW), which
// produces exactly our LDS row stride of LDP=33 floats. Tracked by TENSORcnt.
__device__ __forceinline__ void tdm_load_rows(const float* gsrc, unsigned lds_byte, int rows) {
  unsigned long long ga = (unsigned long long)(size_t)gsrc;
  u32x4 g0;
  g0[0] = 1u;                                    // count=1 (valid), user mode
  g0[1] = lds_byte;                              // LDS byte address
  g0[2] = (unsigned)(ga & 0xFFFFFFFFu);          // global_addr[31:0]
  g0[3] = (unsigned)((ga >> 32) & 0x01FFFFFFu)   // global_addr[56:32]
          | (2u << 30);                          // descriptor type = 2 ("image")
  i32x8 g1;
  g1[0] = (int)((2u << 16)                       // data_size = 4 bytes
                | (1u << 20)                     // pad_enable
                | (4u << 22));                   // pad_interval = 32 DWORDs (amount=1 DW)
  g1[1] = (int)(32u << 16);                      // tensor_dim0 = 32   (bits 79:48, low half)
  g1[2] = (int)(((unsigned)rows) << 16);         // tensor_dim1 = rows (bits 111:80, low half)
  g1[3] = (int)(32u << 16);                      // tile_dim0 = 32     (bits 127:112)
  g1[4] = (int)rows;                             // tile_dim1 = rows; tile_dim2 = 0
  g1[5] = 32;                                    // tensor_dim0_stride = 32 (bits 207:160)
  g1[6] = 0;                                     // tensor_dim1_stride = 0 (unused, 2D)
  g1[7] = 0;
  i32x4 gz4 = {0, 0, 0, 0};
#if defined(__clang_major__) && (__clang_major__ >= 23)
  i32x8 gz8 = {0, 0, 0, 0, 0, 0, 0, 0};
  __builtin_amdgcn_tensor_load_to_lds(g0, g1, gz4, gz4, gz8, 0);
#else
  __builtin_amdgcn_tensor_load_to_lds(g0, g1, gz4, gz4, 0);
#endif
}

__device__ __forceinline__ void wait_tensorcnt_le1() {
#if __has_builtin(__builtin_amdgcn_s_wait_tensorcnt)
  __builtin_amdgcn_s_wait_tensorcnt(1);
#else
  asm volatile("s_wait_tensorcnt 0x1" ::: "memory");
#endif
}
__device__ __forceinline__ void wait_tensorcnt_le0() {
#if __has_builtin(__builtin_amdgcn_s_wait_tensorcnt)
  __builtin_amdgcn_s_wait_tensorcnt(0);
#else
  asm volatile("s_wait_tensorcnt 0x0" ::: "memory");
#endif
}
#endif  // USE_TDM

// -----------------------------------------------------------------------------
// Phase 1: per-chunk cumulative transition matrix P_c (WMMA) and local scan v_c
// grid = NCHUNKS, block = 64 (2 waves). Wave w owns output column block w.
// A_t tiles are double-buffered in LDS and fed by the Tensor Data Mover.
// -----------------------------------------------------------------------------
__global__ __launch_bounds__(64) void pscan_phase1(
    const float* __restrict__ A, const float* __restrict__ X,
    float* __restrict__ Pc, float* __restrict__ vc) {
  __shared__ float sA[2][D_ * LDP];  // double-buffered staged A_t
  __shared__ float sP[D_ * LDP];     // running product
  __shared__ float sV[D_];           // running local scan vector

  const int chunk = blockIdx.x;
  const int s = chunk / NC;
  const int c = chunk % NC;
  const int b = s / H_;
  const int h = s % H_;

  const int tid  = threadIdx.x;
  const int wave = tid >> 5;       // 0..1
  const int lane = tid & 31;
  const int lhi  = lane >> 4;      // K half-select per WMMA f32 layout
  const int llo  = lane & 15;
  const int jw   = wave;           // N column block (0 or 1)

  // P = I, v = 0
  for (int idx = tid; idx < D_ * D_; idx += 64) {
    int r = idx >> 5, cc = idx & 31;
    sP[r * LDP + cc] = (r == cc) ? 1.0f : 0.0f;
  }
  if (tid < D_) sV[tid] = 0.0f;
  __syncthreads();

  const long long aBase = ((long long)b * L_ * H_ + h) * (long long)(D_ * D_);
  const long long xBase = ((long long)b * L_ * H_ + h) * (long long)D_;
  const float* Ac = A + aBase + (long long)(c * CH) * ASTRIDE;  // chunk's first tile

#if USE_TDM
  // Preload t=0 tile: each wave DMA's its own 16-row half.
  tdm_load_rows(Ac + wave * 16 * D_,
                (unsigned)(size_t)&sA[0][wave * 16 * LDP], 16);
#endif

  for (int tt = 0; tt < CH; ++tt) {
    const int t = c * CH + tt;
    const float* At = Ac + (long long)tt * ASTRIDE;
    const int cur = tt & 1;

#if USE_TDM
    if (tt + 1 < CH) {
      tdm_load_rows(At + ASTRIDE + wave * 16 * D_,
                    (unsigned)(size_t)&sA[cur ^ 1][wave * 16 * LDP], 16);
      wait_tensorcnt_le1();   // oldest (current tile) complete; next stays in flight
    } else {
      wait_tensorcnt_le0();
    }
#else
    // Fallback: coalesced stage of A_t (1024 floats, 64 threads x 16)
    #pragma unroll
    for (int k = 0; k < 16; ++k) {
      int idx = tid + k * 64;
      sA[cur][(idx >> 5) * LDP + (idx & 31)] = At[idx];
    }
    if (tt + 1 < CH) __builtin_prefetch(At + ASTRIDE + tid * 16, 0, 0);
#endif
    __syncthreads();

    // B fragments: columns [16*jw, 16*jw+16) of P, all 8 K-blocks (K=4 each)
    v2f bfrag[8];
    #pragma unroll
    for (int kk = 0; kk < 8; ++kk) {
      int krow = 4 * kk + 2 * lhi;
      bfrag[kk].x = sP[krow * LDP + jw * 16 + llo];
      bfrag[kk].y = sP[(krow + 1) * LDP + jw * 16 + llo];
    }

    v8f acc0 = {0.f, 0.f, 0.f, 0.f, 0.f, 0.f, 0.f, 0.f}; // rows 0..15
    v8f acc1 = {0.f, 0.f, 0.f, 0.f, 0.f, 0.f, 0.f, 0.f}; // rows 16..31
    #pragma unroll
    for (int kk = 0; kk < 8; ++kk) {
      int kcol = 4 * kk + 2 * lhi;
      v2f a0, a1;
      a0.x = sA[cur][llo * LDP + kcol];
      a0.y = sA[cur][llo * LDP + kcol + 1];
      a1.x = sA[cur][(16 + llo) * LDP + kcol];
      a1.y = sA[cur][(16 + llo) * LDP + kcol + 1];
      acc0 = wmma_f32_16x16x4(a0, bfrag[kk], acc0);
      acc1 = wmma_f32_16x16x4(a1, bfrag[kk], acc1);
    }

    // v_new = A_t @ v + x_t (wave 0, lane r == row r), reads before barrier
    float vnew = 0.0f;
    if (wave == 0) {
      float acc = X[xBase + (long long)t * (H_ * D_) + lane];
      #pragma unroll 8
      for (int j = 0; j < D_; ++j) acc += sA[cur][lane * LDP + j] * sV[j];
      vnew = acc;
    }
    __syncthreads();  // all reads of sP/sV complete

    // Write back P_new in C/D layout: VGPR r -> row (r + 8*lhi) within tile
    #pragma unroll
    for (int r = 0; r < 8; ++r) {
      sP[(r + 8 * lhi) * LDP + jw * 16 + llo]      = acc0[r];
      sP[(16 + r + 8 * lhi) * LDP + jw * 16 + llo] = acc1[r];
    }
    if (wave == 0) sV[lane] = vnew;
    __syncthreads();
  }

  // Emit chunk summary
  float* Pout = Pc + (long long)chunk * (D_ * D_);
  for (int idx = tid; idx < D_ * D_; idx += 64) {
    Pout[idx] = sP[(idx >> 5) * LDP + (idx & 31)];
  }
  if (tid < D_) vc[chunk * D_ + tid] = sV[tid];
}

// -----------------------------------------------------------------------------
// Phase 2: sequential combine across chunk summaries; record incoming states.
// grid = NSEQ, block = 32 (one wave per sequence).
// -----------------------------------------------------------------------------
__global__ __launch_bounds__(32) void pscan_phase2(
    const float* __restrict__ Pc, const float* __restrict__ vc,
    float* __restrict__ yin) {
  __shared__ float sy[D_];
  const int s = blockIdx.x;
  const int lane = threadIdx.x;

  sy[lane] = 0.0f;
  __syncthreads();

  for (int c = 0; c < NC; ++c) {
    const int chunk = s * NC + c;
    yin[chunk * D_ + lane] = sy[lane];
    const float* P = Pc + (long long)chunk * (D_ * D_);
    float acc = vc[chunk * D_ + lane];
    #pragma unroll 8
    for (int j = 0; j < D_; ++j) acc += P[lane * D_ + j] * sy[j];
    __syncthreads();
    sy[lane] = acc;
    __syncthreads();
  }
}

// -----------------------------------------------------------------------------
// Phase 3: replay exact recurrence within each chunk from its incoming state.
// grid = NCHUNKS, block = 32 (one wave). TDM double-buffers the A_t tiles.
// -----------------------------------------------------------------------------
__global__ __launch_bounds__(32) void pscan_phase3(
    const float* __restrict__ A, const float* __restrict__ X,
    const float* __restrict__ yin, float* __restrict__ Y) {
  __shared__ float sA[2][D_ * LDP];
  __shared__ float sy[D_];

  const int chunk = blockIdx.x;
  const int s = chunk / NC;
  const int c = chunk % NC;
  const int b = s / H_;
  const int h = s % H_;
  const int lane = threadIdx.x;

  sy[lane] = yin[chunk * D_ + lane];
  __syncthreads();

  const long long aBase = ((long long)b * L_ * H_ + h) * (long long)(D_ * D_);
  const long long xBase = ((long long)b * L_ * H_ + h) * (long long)D_;
  const float* Ac = A + aBase + (long long)(c * CH) * ASTRIDE;

#if USE_TDM
  tdm_load_rows(Ac, (unsigned)(size_t)&sA[0][0], 32);
#endif

  for (int tt = 0; tt < CH; ++tt) {
    const int t = c * CH + tt;
    const float* At = Ac + (long long)tt * ASTRIDE;
    const int cur = tt & 1;

#if USE_TDM
    if (tt + 1 < CH) {
      tdm_load_rows(At + ASTRIDE, (unsigned)(size_t)&sA[cur ^ 1][0], 32);
      wait_tensorcnt_le1();
    } else {
      wait_tensorcnt_le0();
    }
#else
    #pragma unroll
    for (int k = 0; k < 32; ++k) {
      int idx = lane + k * 32;
      sA[cur][(idx >> 5) * LDP + (idx & 31)] = At[idx];
    }
    if (tt + 1 < CH) __builtin_prefetch(At + ASTRIDE + lane * 32, 0, 0);
#endif
    __syncthreads();

    float acc = X[xBase + (long long)t * (H_ * D_) + lane];
    #pragma unroll 8
    for (int j = 0; j < D_; ++j) acc += sA[cur][lane * LDP + j] * sy[j];
    __syncthreads();
    sy[lane] = acc;
    Y[xBase + (long long)t * (H_ * D_) + lane] = acc;
    __syncthreads();
  }
}

// -----------------------------------------------------------------------------
extern "C" void kernel_launch(void* const* d_in, const int* in_sizes, int n_in,
                              void* d_out, int out_size, void* d_ws, size_t ws_size,
                              hipStream_t stream) {
  const float* A = (const float*)d_in[0];
  const float* X = (const float*)d_in[1];
  float* Y = (float*)d_out;

  // Workspace layout (floats): P_c [2048][32][32], v_c [2048][32], yin [2048][32]
  float* ws  = (float*)d_ws;
  float* Pc  = ws;
  float* vc  = Pc + (long long)NCHUNKS * D_ * D_;
  float* yin = vc + (long long)NCHUNKS * D_;

  pscan_phase1<<<NCHUNKS, 64, 0, stream>>>(A, X, Pc, vc);
  pscan_phase2<<<NSEQ, 32, 0, stream>>>(Pc, vc, yin);
  pscan_phase3<<<NCHUNKS, 32, 0, stream>>>(A, X, yin, Y);
}